// GraphConvolution_77403900609155
// MI455X (gfx1250) — compile-verified
//
#include <hip/hip_runtime.h>
#include <hip/hip_bf16.h>

#define N_NODES 100000
#define N_EDGES 3200000
#define D_IN    256
#define D_OUT   128
#define KC      64                    // K-chunk staged in LDS (64x128 f32 = 32 KB)
#define ROW_TILES (N_NODES / 16)      // 6250 full 16-row tiles (100000 % 16 == 0)

typedef float v2f __attribute__((ext_vector_type(2)));
typedef float v8f __attribute__((ext_vector_type(8)));
typedef int   v4i __attribute__((ext_vector_type(4)));

// Address-space-qualified v4i pointers for the async DMA builtin
// (clang prints AS1 as "__device__", AS3 as "__shared__" in HIP diagnostics).
typedef __attribute__((address_space(1))) v4i g_v4i;
typedef __attribute__((address_space(3))) v4i l_v4i;

// --- CDNA5 async global->LDS (ASYNCcnt-tracked DMA), with safe fallback -----
#if defined(__has_builtin)
#if __has_builtin(__builtin_amdgcn_global_load_async_to_lds_b128)
#define HAVE_ASYNC_LDS 1
#endif
#endif
#ifndef HAVE_ASYNC_LDS
#define HAVE_ASYNC_LDS 0
#endif

__device__ __forceinline__ void stage_chunk(const float4* __restrict__ wg,
                                            float4* wl, int tid) {
#if HAVE_ASYNC_LDS
    #pragma unroll
    for (int i = 0; i < 8; ++i) {                 // 8 x b128 per thread
        const int idx = tid + i * 256;
        __builtin_amdgcn_global_load_async_to_lds_b128(
            (g_v4i*)(wg + idx),
            (l_v4i*)(wl + idx),
            /*imm offset=*/0, /*cpol=*/0);
    }
#else
    #pragma unroll
    for (int i = 0; i < 8; ++i) {
        const int idx = tid + i * 256;
        wl[idx] = wg[idx];
    }
#endif
}

__device__ __forceinline__ void wait_stage() {
#if HAVE_ASYNC_LDS
#if __has_builtin(__builtin_amdgcn_s_wait_asynccnt)
    __builtin_amdgcn_s_wait_asynccnt(0);
#else
    asm volatile("s_wait_asynccnt 0x0" ::: "memory");
#endif
#endif
    __syncthreads();
}

// ---------------------------------------------------------------------------
// Zero the output (harness poisons d_out with 0xAA before timing).
// ---------------------------------------------------------------------------
__global__ void gcn_zero_f4(float4* __restrict__ p, int n4) {
    int i = blockIdx.x * blockDim.x + threadIdx.x;
    if (i < n4) p[i] = make_float4(0.f, 0.f, 0.f, 0.f);
}

// ---------------------------------------------------------------------------
// support = x @ W  via V_WMMA_F32_16X16X4_F32 (exact fp32 matrix math).
// Block = 256 threads = 8 waves (wave32). Wave w computes rows
// [rowTile*16, rowTile*16+16) x all 128 cols (8 column tiles, 8 v8f accs).
// W is double-buffered through LDS in 64x128 chunks via async DMA; chunk i+1
// streams in while WMMAs consume chunk i. x is read exactly once -> NT loads.
//
// ISA 7.12.2 layouts (wave32):
//   A 16x4 f32 : lane<16 -> {A[m][k+0], A[m][k+1]},  lane>=16 -> {A[m][k+2], A[m][k+3]},  m = lane&15
//   B 4x16 f32 : lane<16 -> {B[k+0][n], B[k+1][n]},  lane>=16 -> {B[k+2][n], B[k+3][n]},  n = lane&15
//   C/D 16x16  : vgpr j   : lane<16 -> D[j][lane],   lane>=16 -> D[j+8][lane-16]
// ---------------------------------------------------------------------------
__global__ void __launch_bounds__(256)
gcn_gemm_wmma(const float* __restrict__ X, const float* __restrict__ W,
              float* __restrict__ S) {
    __shared__ float wlds[2][KC * D_OUT];         // 2 x 32 KB double buffer

    const int tid    = threadIdx.x;
    const int lane   = tid & 31;
    const int wave   = tid >> 5;
    const int rowTile = blockIdx.x * 8 + wave;
    const bool active = (rowTile < ROW_TILES);

    const int m     = lane & 15;                  // M (A) / N (B) index
    const int khalf = (lane >> 4) << 1;           // 0 or 2 per half-wave

    v8f acc[8] = {};                              // 8 col-tiles x 8 VGPRs

    const float* xrow = X + (size_t)(rowTile * 16 + m) * D_IN;

    // Prologue: stage chunk 0
    stage_chunk(reinterpret_cast<const float4*>(W), (float4*)wlds[0], tid);
    wait_stage();

    int cur = 0;
    for (int kc = 0; kc < D_IN; kc += KC) {
        // Prefetch next chunk into the other buffer while computing this one.
        const bool more = (kc + KC) < D_IN;
        if (more) {
            stage_chunk(reinterpret_cast<const float4*>(W + (size_t)(kc + KC) * D_OUT),
                        (float4*)wlds[cur ^ 1], tid);
        }

        if (active) {
            const float* wb = wlds[cur];
            #pragma unroll
            for (int kk = 0; kk < KC; kk += 4) {
                const int k = kc + kk + khalf;
                const v2f a = __builtin_nontemporal_load(
                    reinterpret_cast<const v2f*>(xrow + k));   // NT: x streamed once
                #pragma unroll
                for (int ct = 0; ct < 8; ++ct) {
                    const int n = ct * 16 + m;
                    v2f b;
                    b.x = wb[(kk + khalf)     * D_OUT + n];
                    b.y = wb[(kk + khalf + 1) * D_OUT + n];
                    acc[ct] = __builtin_amdgcn_wmma_f32_16x16x4_f32(
                        /*neg_a=*/false, a, /*neg_b=*/false, b,
                        /*c_mod=*/(short)0, acc[ct],
                        /*reuse_a=*/false, /*reuse_b=*/false);
                }
            }
        }

        if (more) wait_stage();     // next buffer ready; all waves done with cur
        cur ^= 1;
    }

    if (active) {
        const int rbase = rowTile * 16 + ((lane >> 4) ? 8 : 0);
        #pragma unroll
        for (int ct = 0; ct < 8; ++ct) {
            const int col = ct * 16 + m;
            #pragma unroll
            for (int j = 0; j < 8; ++j) {
                S[(size_t)(rbase + j) * D_OUT + col] = acc[ct][j];
            }
        }
    }
}

// ---------------------------------------------------------------------------
// out[dst] += w_e * support[src]  — one wave per edge.
// Lane l handles floats [4l, 4l+4): coalesced 512B float4 gather (support and
// out stay L2-resident: 51MB + 51MB << 192MB) + 4 non-returning f32 atomics.
// Edge arrays are read exactly once -> NT loads so they don't evict the
// reused working set from L2.
// ---------------------------------------------------------------------------
__global__ void __launch_bounds__(256)
gcn_spmm_scatter(const float* __restrict__ support,
                 const int* __restrict__ esrc, const int* __restrict__ edst,
                 const float* __restrict__ ew, float* __restrict__ out,
                 int nEdges) {
    const int lane = threadIdx.x & 31;
    const int wave = threadIdx.x >> 5;
    const int e = blockIdx.x * 8 + wave;
    if (e >= nEdges) return;

    const int   s  = __builtin_nontemporal_load(esrc + e);
    const int   d  = __builtin_nontemporal_load(edst + e);
    const float wt = __builtin_nontemporal_load(ew + e);

    const float4 v =
        reinterpret_cast<const float4*>(support + (size_t)s * D_OUT)[lane];

    float* o = out + (size_t)d * D_OUT + lane * 4;
    atomicAdd(o + 0, v.x * wt);
    atomicAdd(o + 1, v.y * wt);
    atomicAdd(o + 2, v.z * wt);
    atomicAdd(o + 3, v.w * wt);
}

// ---------------------------------------------------------------------------
extern "C" void kernel_launch(void* const* d_in, const int* in_sizes, int n_in,
                              void* d_out, int out_size, void* d_ws, size_t ws_size,
                              hipStream_t stream) {
    const float* x    = (const float*)d_in[0];
    const int*   esrc = (const int*)  d_in[1];
    const int*   edst = (const int*)  d_in[2];
    const float* ew   = (const float*)d_in[3];
    const float* W    = (const float*)d_in[4];
    float*       out  = (float*)d_out;
    float*       supp = (float*)d_ws;            // needs N_NODES*D_OUT*4 = 51.2 MB

    // 1) zero the accumulation target
    const int n4 = N_NODES * D_OUT / 4;
    gcn_zero_f4<<<(n4 + 255) / 256, 256, 0, stream>>>((float4*)out, n4);

    // 2) dense GEMM on the matrix cores (fp32 WMMA, async-DMA double buffering)
    const int gemmBlocks = (ROW_TILES + 7) / 8;  // 782
    gcn_gemm_wmma<<<gemmBlocks, 256, 0, stream>>>(x, W, supp);

    // 3) edge gather + scatter-add
    const int spmmBlocks = (N_EDGES + 7) / 8;    // 400000
    gcn_spmm_scatter<<<spmmBlocks, 256, 0, stream>>>(supp, esrc, edst, ew, out,
                                                     N_EDGES);
}